// PrivateGraphSAGE_5927054868542
// MI455X (gfx1250) — compile-verified
//
#include <hip/hip_runtime.h>

#define N_NODES 100000
#define N_EDGES 1200000
#define D 64
#define EPS 1e-12f
#define NTILES (N_NODES / 16)   // 6250 exactly

typedef float v2f __attribute__((ext_vector_type(2)));
typedef float v8f __attribute__((ext_vector_type(8)));

// ---------------------------------------------------------------------------
// Helper: sum of squares of 32 consecutive floats (half of a 64-wide row)
// ---------------------------------------------------------------------------
__device__ __forceinline__ float half_row_sumsq(const float* __restrict__ p) {
    float ss = 0.0f;
#pragma unroll
    for (int j = 0; j < 8; ++j) {
        const float4 v = ((const float4*)p)[j];
        ss += v.x * v.x + v.y * v.y + v.z * v.z + v.w * v.w;
    }
    return ss;
}

// ---------------------------------------------------------------------------
// prep: per-row L2 norm of X; write xn to XN (gather source) and AGG (init),
// store ||x|| in NRM.  One wave of 32 lanes handles 16 rows (lane = m + 16*hi).
// ---------------------------------------------------------------------------
__global__ __launch_bounds__(256)
void prep_kernel(const float* __restrict__ X,
                 float* __restrict__ XN,
                 float* __restrict__ AGG,
                 float* __restrict__ NRM) {
    const int wave = blockIdx.x * (blockDim.x >> 5) + (threadIdx.x >> 5);
    if (wave >= NTILES) return;
    const int lane = threadIdx.x & 31;
    const int m = lane & 15, hi = lane >> 4;
    const long long row = (long long)wave * 16 + m;

    const float* xr = X + row * D + hi * 32;
    float ss = half_row_sumsq(xr);
    ss += __shfl_xor(ss, 16, 32);
    const float n   = sqrtf(ss);
    const float inv = 1.0f / fmaxf(n, EPS);
    if (hi == 0) NRM[row] = n;

    float* xnw = XN  + row * D + hi * 32;
    float* agw = AGG + row * D + hi * 32;
#pragma unroll
    for (int j = 0; j < 8; ++j) {
        float4 v = ((const float4*)xr)[j];
        v.x *= inv; v.y *= inv; v.z *= inv; v.w *= inv;
        ((float4*)xnw)[j] = v;
        ((float4*)agw)[j] = v;
    }
}

// ---------------------------------------------------------------------------
// scatter: AGG[dst] += XN[src] over all edges. One wave per edge, 2 floats
// per lane via fp32 global atomics (L2-resident; hardware fp32 atomic add).
// ---------------------------------------------------------------------------
__global__ __launch_bounds__(256)
void scatter_kernel(const float* __restrict__ XN,
                    const int* __restrict__ ei,
                    float* __restrict__ AGG) {
    const long long t = (long long)blockIdx.x * blockDim.x + threadIdx.x;
    const int e = (int)(t >> 5);
    if (e >= N_EDGES) return;
    const int lane = threadIdx.x & 31;
    const int s = ei[e];              // broadcast load (all lanes same addr)
    const int d = ei[N_EDGES + e];
    const int f = lane * 2;
    const v2f v = *(const v2f*)(XN + (long long)s * D + f);
    float* a = AGG + (long long)d * D + f;
    unsafeAtomicAdd(a,     v.x);
    unsafeAtomicAdd(a + 1, v.y);
}

// ---------------------------------------------------------------------------
// update: per 16-node tile (one wave):
//   s[m]  = scale * ||x_m|| / max(||agg_m||, eps)
//   out   = (agg * s) @ Wl^T + bl + x @ Wr^T    via V_WMMA_F32_16X16X4_F32
//   optional ReLU, store.
// Weights staged in LDS per 256-thread block (8 waves -> 128 nodes).
// ---------------------------------------------------------------------------
__global__ __launch_bounds__(256)
void update_kernel(const float* __restrict__ AGG,
                   const float* __restrict__ NRM,
                   const float* __restrict__ X,
                   const float* __restrict__ Wl,
                   const float* __restrict__ bl,
                   const float* __restrict__ Wr,
                   const float* __restrict__ scalep,
                   float* __restrict__ OUT,
                   const int do_relu) {
    __shared__ __align__(16) float sWl[D * D];
    __shared__ __align__(16) float sWr[D * D];
    __shared__ __align__(16) float sbl[D];

    for (int i = threadIdx.x; i < D * D; i += blockDim.x) {
        sWl[i] = Wl[i];
        sWr[i] = Wr[i];
    }
    if (threadIdx.x < D) sbl[threadIdx.x] = bl[threadIdx.x];
    __syncthreads();

    const int tile = blockIdx.x * (blockDim.x >> 5) + (threadIdx.x >> 5);
    if (tile >= NTILES) return;   // wave-uniform; EXEC stays all-ones below

    const int lane = threadIdx.x & 31;
    const int m = lane & 15, hi = lane >> 4;
    const long long row = (long long)tile * 16 + m;
    const float scale = scalep[0];

    // message scale factor for row m (shared by both lanes holding row m)
    float ss = half_row_sumsq(AGG + row * D + hi * 32);
    ss += __shfl_xor(ss, 16, 32);
    const float srow = scale * NRM[row] / fmaxf(sqrtf(ss), EPS);

    const float* aRow = AGG + row * D;
    const float* xRow = X   + row * D;

    // C/D layout: acc[t][r] = out[M = r + 8*hi][N = t*16 + m]; bias depends on N only
    v8f acc[4];
#pragma unroll
    for (int t = 0; t < 4; ++t) {
        const float bt = sbl[t * 16 + m];
#pragma unroll
        for (int r = 0; r < 8; ++r) acc[t][r] = bt;
    }

    // K loop: 16 steps of K=4; A frag = float2 at col (k0 + 2*hi);
    // B frag = float2 of W[(t*16+m)*64 + k0 + 2*hi] (row-major W == B^T layout)
#pragma unroll 4
    for (int k0 = 0; k0 < D; k0 += 4) {
        const int kk = k0 + 2 * hi;
        v2f amsg = *(const v2f*)(aRow + kk);
        amsg.x *= srow;
        amsg.y *= srow;
        const v2f ax = *(const v2f*)(xRow + kk);
#pragma unroll
        for (int t = 0; t < 4; ++t) {
            const int n = t * 16 + m;
            const v2f bfl = *(const v2f*)(sWl + n * D + kk);
            const v2f bfr = *(const v2f*)(sWr + n * D + kk);
            acc[t] = __builtin_amdgcn_wmma_f32_16x16x4_f32(
                false, amsg, false, bfl, (short)0, acc[t], false, false);
            acc[t] = __builtin_amdgcn_wmma_f32_16x16x4_f32(
                false, ax,   false, bfr, (short)0, acc[t], false, false);
        }
    }

    // store: lanes 0..15 of each half write 16 consecutive floats per (t, r)
#pragma unroll
    for (int t = 0; t < 4; ++t) {
#pragma unroll
        for (int r = 0; r < 8; ++r) {
            float v = acc[t][r];
            if (do_relu) v = fmaxf(v, 0.0f);
            OUT[((long long)tile * 16 + r + 8 * hi) * D + t * 16 + m] = v;
        }
    }
}

// ---------------------------------------------------------------------------
// finalize: in-place row-wise L2 normalize of Y (the final layer output)
// ---------------------------------------------------------------------------
__global__ __launch_bounds__(256)
void finalize_kernel(float* __restrict__ Y) {
    const int wave = blockIdx.x * (blockDim.x >> 5) + (threadIdx.x >> 5);
    if (wave >= NTILES) return;
    const int lane = threadIdx.x & 31;
    const int m = lane & 15, hi = lane >> 4;
    float* yr = Y + ((long long)wave * 16 + m) * D + hi * 32;

    float4 v[8];
    float ss = 0.0f;
#pragma unroll
    for (int j = 0; j < 8; ++j) {
        v[j] = ((const float4*)yr)[j];
        ss += v[j].x * v[j].x + v[j].y * v[j].y + v[j].z * v[j].z + v[j].w * v[j].w;
    }
    ss += __shfl_xor(ss, 16, 32);
    const float inv = 1.0f / fmaxf(sqrtf(ss), EPS);
#pragma unroll
    for (int j = 0; j < 8; ++j) {
        v[j].x *= inv; v[j].y *= inv; v[j].z *= inv; v[j].w *= inv;
        ((float4*)yr)[j] = v[j];
    }
}

// ---------------------------------------------------------------------------
extern "C" void kernel_launch(void* const* d_in, const int* in_sizes, int n_in,
                              void* d_out, int out_size, void* d_ws, size_t ws_size,
                              hipStream_t stream) {
    const float* x   = (const float*)d_in[0];
    const float* Wl0 = (const float*)d_in[1];
    const float* bl0 = (const float*)d_in[2];
    const float* Wr0 = (const float*)d_in[3];
    const float* sc0 = (const float*)d_in[4];
    const float* Wl1 = (const float*)d_in[5];
    const float* bl1 = (const float*)d_in[6];
    const float* Wr1 = (const float*)d_in[7];
    const float* sc1 = (const float*)d_in[8];
    const int*   ei  = (const int*)d_in[9];

    float* ws  = (float*)d_ws;
    float* XN  = ws;                              // N*64
    float* AGG = XN  + (size_t)N_NODES * D;       // N*64
    float* H   = AGG + (size_t)N_NODES * D;       // N*64
    float* NRM = H   + (size_t)N_NODES * D;       // N
    float* OUT = (float*)d_out;

    const dim3 blk(256);
    const int nodeBlocks    = (NTILES + 7) / 8;          // 8 waves/block
    const int scatterBlocks = (N_EDGES * 32) / 256;      // 1 wave/edge

    // ---- Layer 0 ----
    prep_kernel<<<nodeBlocks, blk, 0, stream>>>(x, XN, AGG, NRM);
    scatter_kernel<<<scatterBlocks, blk, 0, stream>>>(XN, ei, AGG);
    update_kernel<<<nodeBlocks, blk, 0, stream>>>(AGG, NRM, x, Wl0, bl0, Wr0, sc0, H, 1);

    // ---- Layer 1 ----
    prep_kernel<<<nodeBlocks, blk, 0, stream>>>(H, XN, AGG, NRM);
    scatter_kernel<<<scatterBlocks, blk, 0, stream>>>(XN, ei, AGG);
    update_kernel<<<nodeBlocks, blk, 0, stream>>>(AGG, NRM, H, Wl1, bl1, Wr1, sc1, OUT, 0);

    // ---- final l2norm ----
    finalize_kernel<<<nodeBlocks, blk, 0, stream>>>(OUT);
}